// ChamferLoss_25194278158453
// MI455X (gfx1250) — compile-verified
//
#include <hip/hip_runtime.h>
#include <math.h>

#define BATCH 128
#define NN    256
#define CCLS  5
#define CCHG  3
#define SMAX  300

typedef float v2f __attribute__((ext_vector_type(2)));
typedef float v8f __attribute__((ext_vector_type(8)));

// ---------------------------------------------------------------------------
// Kernel 0: zero the scalar output (harness poisons d_out with 0xAA).
// ---------------------------------------------------------------------------
__global__ void zero_kernel(float* __restrict__ out) {
    if (threadIdx.x == 0 && blockIdx.x == 0) out[0] = 0.0f;
}

// ---------------------------------------------------------------------------
// Kernel 1: per-batch 256x256 pairwise cost + chamfer row/col mins.
// One workgroup (256 threads = 8 wave32) per batch element.
// Each wave owns 16-row blocks and sweeps all 16 column tiles, so all
// A-side WMMA operands and row scalars are hoisted out of the inner loop.
// Bilinear cost pieces run on V_WMMA_F32_16X16X4_F32.
// ---------------------------------------------------------------------------
__global__ __launch_bounds__(256) void chamfer_pair_kernel(
    const int*   __restrict__ p_cls,      // (B,N) particle_class
    const int*   __restrict__ p_chg,      // (B,N) particle_charge
    const float* __restrict__ cls_logits, // (B,N,5) pflow_class_logits
    const float* __restrict__ chg_logits, // (B,N,3) pflow_charge_logits
    const float* __restrict__ p_pos,      // (B,N,3)
    const float* __restrict__ q_pos,      // (B,N,3)
    const float* __restrict__ p_mom,      // (B,N,3)
    const float* __restrict__ q_mom,      // (B,N,3)
    const float* __restrict__ p_en,       // (B,N)
    const float* __restrict__ q_en,       // (B,N)
    float*       __restrict__ out)        // scalar accumulator
{
    __shared__ float s_clsn[NN][CCLS];   // -log softmax of pflow class logits
    __shared__ float s_chgn[NN][CCHG];   // -log softmax of pflow charge logits
    __shared__ int   s_pc[NN];           // particle class labels
    __shared__ int   s_pg[NN];           // particle charge labels
    __shared__ float s_pp[NN][3];        // particle pos
    __shared__ float s_qp[NN][3];        // pflow pos
    __shared__ float s_pm[NN][3];        // particle mom
    __shared__ float s_qm[NN][3];        // pflow mom
    __shared__ float s_pp2[NN], s_qp2[NN], s_pm2[NN], s_qm2[NN]; // squared norms
    __shared__ float s_pe[NN], s_qe[NN]; // energies
    __shared__ int   s_minp[NN];         // min over j per particle i (float bits)
    __shared__ int   s_minq[NN];         // min over i per pflow j (float bits)
    __shared__ float s_red[256];

    const int    b   = blockIdx.x;
    const int    tid = threadIdx.x;      // 0..255, doubles as row index j
    const size_t rb  = (size_t)b * NN + tid;

    // ---- stage everything for this batch into LDS --------------------------
    {
        const float* cl = cls_logits + rb * CCLS;
        float x0 = cl[0], x1 = cl[1], x2 = cl[2], x3 = cl[3], x4 = cl[4];
        float m  = fmaxf(fmaxf(fmaxf(x0, x1), fmaxf(x2, x3)), x4);
        float s  = __expf(x0 - m) + __expf(x1 - m) + __expf(x2 - m)
                 + __expf(x3 - m) + __expf(x4 - m);
        float lse = m + __logf(s);
        s_clsn[tid][0] = lse - x0; s_clsn[tid][1] = lse - x1;
        s_clsn[tid][2] = lse - x2; s_clsn[tid][3] = lse - x3;
        s_clsn[tid][4] = lse - x4;
    }
    {
        const float* cg = chg_logits + rb * CCHG;
        float x0 = cg[0], x1 = cg[1], x2 = cg[2];
        float m  = fmaxf(fmaxf(x0, x1), x2);
        float s  = __expf(x0 - m) + __expf(x1 - m) + __expf(x2 - m);
        float lse = m + __logf(s);
        s_chgn[tid][0] = lse - x0; s_chgn[tid][1] = lse - x1;
        s_chgn[tid][2] = lse - x2;
    }
    s_pc[tid] = p_cls[rb];
    s_pg[tid] = p_chg[rb];
    {
        float a0 = p_pos[rb*3+0], a1 = p_pos[rb*3+1], a2 = p_pos[rb*3+2];
        s_pp[tid][0] = a0; s_pp[tid][1] = a1; s_pp[tid][2] = a2;
        s_pp2[tid] = a0*a0 + a1*a1 + a2*a2;
        float b0 = q_pos[rb*3+0], b1 = q_pos[rb*3+1], b2 = q_pos[rb*3+2];
        s_qp[tid][0] = b0; s_qp[tid][1] = b1; s_qp[tid][2] = b2;
        s_qp2[tid] = b0*b0 + b1*b1 + b2*b2;
        float c0 = p_mom[rb*3+0], c1 = p_mom[rb*3+1], c2 = p_mom[rb*3+2];
        s_pm[tid][0] = c0; s_pm[tid][1] = c1; s_pm[tid][2] = c2;
        s_pm2[tid] = c0*c0 + c1*c1 + c2*c2;
        float d0 = q_mom[rb*3+0], d1 = q_mom[rb*3+1], d2 = q_mom[rb*3+2];
        s_qm[tid][0] = d0; s_qm[tid][1] = d1; s_qm[tid][2] = d2;
        s_qm2[tid] = d0*d0 + d1*d1 + d2*d2;
    }
    s_pe[tid] = p_en[rb];
    s_qe[tid] = q_en[rb];
    s_minp[tid] = 0x7F7FFFFF;   // +FLT_MAX bits; all costs are >= 0
    s_minq[tid] = 0x7F7FFFFF;
    __syncthreads();

    // ---- row-block loop: each wave owns 2 of the 16 row blocks -------------
    const int lane = tid & 31;
    const int wv   = tid >> 5;       // wave id 0..7
    const int h    = lane >> 4;      // half-wave selector (K-split for A/B)
    const int l    = lane & 15;      // M (for A) / N (for B,C,D) index

    for (int rblk = wv; rblk < 16; rblk += 8) {
        const int i0 = rblk << 4;
        const int ia = i0 + l;       // A-matrix row handled by this lane

        // ---- hoisted A-side WMMA operands (constant over all column tiles)
        const int c = s_pc[ia];
        const int g = s_pg[ia];
        v2f a_c1, a_c2, a_g, a_p, a_m;
        a_c1.x = (h ? (c == 2) : (c == 0)) ? 1.0f : 0.0f;  // one-hot, K 0..3
        a_c1.y = (h ? (c == 3) : (c == 1)) ? 1.0f : 0.0f;
        a_c2.x = h ? 0.0f : ((c == 4) ? 1.0f : 0.0f);      // class 4, K slot 0
        a_c2.y = 0.0f;
        a_g.x  = (h ? (g == 2) : (g == 0)) ? 1.0f : 0.0f;  // charge one-hot
        a_g.y  = h ? 0.0f : ((g == 1) ? 1.0f : 0.0f);
        a_p.x  = s_pp[ia][h ? 2 : 0];                      // pos (K=3 padded)
        a_p.y  = h ? 0.0f : s_pp[ia][1];
        a_m.x  = s_pm[ia][h ? 2 : 0];                      // mom (K=3 padded)
        a_m.y  = h ? 0.0f : s_pm[ia][1];

        // ---- hoisted row scalars (C/D layout: row = i0 + 8h + r)
        float r_pp2[8], r_pm2[8], r_pe[8], rowmin[8];
        #pragma unroll
        for (int r = 0; r < 8; ++r) {
            const int i = i0 + (h << 3) + r;
            r_pp2[r] = s_pp2[i];
            r_pm2[r] = s_pm2[i];
            r_pe[r]  = s_pe[i];
            rowmin[r] = 3.402823466e38f;
        }

        // ---- sweep all 16 column tiles -----------------------------------
        for (int j0 = 0; j0 < NN; j0 += 16) {
            const int jb = j0 + l;   // B/C/D column handled by this lane

            v2f bb;
            v8f c_nll = {0.f,0.f,0.f,0.f,0.f,0.f,0.f,0.f};

            bb.x = h ? s_clsn[jb][2] : s_clsn[jb][0];
            bb.y = h ? s_clsn[jb][3] : s_clsn[jb][1];
            c_nll = __builtin_amdgcn_wmma_f32_16x16x4_f32(
                false, a_c1, false, bb, (short)0, c_nll, false, false);

            bb.x = h ? 0.0f : s_clsn[jb][4];
            bb.y = 0.0f;
            c_nll = __builtin_amdgcn_wmma_f32_16x16x4_f32(
                false, a_c2, false, bb, (short)0, c_nll, false, false);

            bb.x = h ? s_chgn[jb][2] : s_chgn[jb][0];
            bb.y = h ? 0.0f : s_chgn[jb][1];
            c_nll = __builtin_amdgcn_wmma_f32_16x16x4_f32(
                false, a_g, false, bb, (short)0, c_nll, false, false);

            v8f c_pos = {0.f,0.f,0.f,0.f,0.f,0.f,0.f,0.f};
            bb.x = s_qp[jb][h ? 2 : 0];
            bb.y = h ? 0.0f : s_qp[jb][1];
            c_pos = __builtin_amdgcn_wmma_f32_16x16x4_f32(
                false, a_p, false, bb, (short)0, c_pos, false, false);

            v8f c_mom = {0.f,0.f,0.f,0.f,0.f,0.f,0.f,0.f};
            bb.x = s_qm[jb][h ? 2 : 0];
            bb.y = h ? 0.0f : s_qm[jb][1];
            c_mom = __builtin_amdgcn_wmma_f32_16x16x4_f32(
                false, a_m, false, bb, (short)0, c_mom, false, false);

            const float qp2 = s_qp2[jb];
            const float qm2 = s_qm2[jb];
            const float qe  = s_qe[jb];
            float colmin = 3.402823466e38f;
            #pragma unroll
            for (int r = 0; r < 8; ++r) {
                float posd = __builtin_amdgcn_sqrtf(
                    fmaxf(0.f, r_pp2[r] + qp2 - 2.f * c_pos[r]));
                float momd = __builtin_amdgcn_sqrtf(
                    fmaxf(0.f, r_pm2[r] + qm2 - 2.f * c_mom[r]));
                float de   = r_pe[r] - qe;
                float cost = c_nll[r] + posd + momd + de * de;
                colmin     = fminf(colmin, cost);
                rowmin[r]  = fminf(rowmin[r], cost);
            }
            atomicMin(&s_minq[jb], __float_as_int(colmin)); // min over i, col j
        }

        // ---- one atomic per row after the full column sweep --------------
        #pragma unroll
        for (int r = 0; r < 8; ++r) {
            const int i = i0 + (h << 3) + r;
            atomicMin(&s_minp[i], __float_as_int(rowmin[r]));
        }
    }
    __syncthreads();

    // ---- per-batch sum of row-mins + col-mins, then global accumulate ------
    s_red[tid] = __int_as_float(s_minp[tid]) + __int_as_float(s_minq[tid]);
    __syncthreads();
    for (int off = 128; off > 0; off >>= 1) {
        if (tid < off) s_red[tid] += s_red[tid + off];
        __syncthreads();
    }
    if (tid == 0) atomicAdd(out, s_red[0] * (1.0f / BATCH));
}

// ---------------------------------------------------------------------------
// Kernel 2: set-size loss. One block per batch; mean over N, log-softmax over
// S_MAX=300, gather at n_particles[b].
// ---------------------------------------------------------------------------
__global__ __launch_bounds__(256) void setsize_kernel(
    const float* __restrict__ ss,     // (B,N,SMAX)
    const int*   __restrict__ npart,  // (B,)
    float*       __restrict__ out)
{
    __shared__ float s_val[512];
    __shared__ float s_red[256];
    const int b   = blockIdx.x;
    const int tid = threadIdx.x;

    for (int s = tid; s < 512; s += 256) {
        if (s < SMAX) {
            const float* base = ss + (size_t)b * NN * SMAX + s;
            float acc = 0.0f;
            for (int n = 0; n < NN; ++n) acc += base[(size_t)n * SMAX];
            s_val[s] = acc * (1.0f / NN);
        } else {
            s_val[s] = -INFINITY;
        }
    }
    __syncthreads();

    // block max over 512 staged values
    s_red[tid] = fmaxf(s_val[tid], s_val[tid + 256]);
    __syncthreads();
    for (int off = 128; off > 0; off >>= 1) {
        if (tid < off) s_red[tid] = fmaxf(s_red[tid], s_red[tid + off]);
        __syncthreads();
    }
    const float mx = s_red[0];
    __syncthreads();

    // block sum of exp(v - mx)
    float e = (tid       < SMAX ? __expf(s_val[tid]       - mx) : 0.0f)
            + (tid + 256 < SMAX ? __expf(s_val[tid + 256] - mx) : 0.0f);
    s_red[tid] = e;
    __syncthreads();
    for (int off = 128; off > 0; off >>= 1) {
        if (tid < off) s_red[tid] += s_red[tid + off];
        __syncthreads();
    }
    if (tid == 0) {
        const float lse = mx + __logf(s_red[0]);
        const int   tgt = npart[b];           // in [0, SMAX)
        const float nll = lse - s_val[tgt];   // -(v - lse)
        atomicAdd(out, nll * (1.0f / BATCH));
    }
}

// ---------------------------------------------------------------------------
extern "C" void kernel_launch(void* const* d_in, const int* in_sizes, int n_in,
                              void* d_out, int out_size, void* d_ws, size_t ws_size,
                              hipStream_t stream) {
    const int*   p_cls      = (const int*)  d_in[0];
    const int*   p_chg      = (const int*)  d_in[1];
    const int*   n_part     = (const int*)  d_in[2];
    const float* cls_logits = (const float*)d_in[3];
    const float* chg_logits = (const float*)d_in[4];
    const float* p_pos      = (const float*)d_in[5];
    const float* q_pos      = (const float*)d_in[6];
    const float* p_mom      = (const float*)d_in[7];
    const float* q_mom      = (const float*)d_in[8];
    const float* p_en       = (const float*)d_in[9];
    const float* q_en       = (const float*)d_in[10];
    const float* setsizes   = (const float*)d_in[11];
    float* out = (float*)d_out;

    zero_kernel<<<1, 1, 0, stream>>>(out);
    chamfer_pair_kernel<<<BATCH, 256, 0, stream>>>(
        p_cls, p_chg, cls_logits, chg_logits,
        p_pos, q_pos, p_mom, q_mom, p_en, q_en, out);
    setsize_kernel<<<BATCH, 256, 0, stream>>>(setsizes, n_part, out);
}